// BGAN_48979807043935
// MI455X (gfx1250) — compile-verified
//
#include <hip/hip_runtime.h>
#include <hip/hip_bf16.h>

// Problem constants (from reference)
#define NN      65536
#define KK      10
#define DD      256
#define CC      40
#define GSTR    272          // gat_out padded 265 -> 272 (multiple of 4)
#define BN_EPS  1e-5f

typedef __attribute__((ext_vector_type(2))) float v2f;
typedef __attribute__((ext_vector_type(8))) float v8f;

__device__ __forceinline__ v8f wmma4(v2f a, v2f b, v8f c) {
    // V_WMMA_F32_16X16X4_F32 : D = A(16x4) x B(4x16) + C(16x16), fp32 exact
    return __builtin_amdgcn_wmma_f32_16x16x4_f32(
        false, a, false, b, (short)0, c, false, false);
}

// ---------------------------------------------------------------------------
// GEMM1: z[n][j] = sum_d h[n][d] * Wfc[j][d]   (z = h @ W_fc^T)
// one wave per 16x64 output strip; K-loop 256/4
// ---------------------------------------------------------------------------
__global__ __launch_bounds__(32) void k_gemm_z(const float* __restrict__ h,
                                               const float* __restrict__ Wfc,
                                               float* __restrict__ z) {
    const int m0   = blockIdx.x * 16;
    const int j0   = blockIdx.y * 64;
    const int lane = threadIdx.x;
    const int lo   = lane & 15;
    const int hi   = lane >> 4;

    v8f acc0 = {}, acc1 = {}, acc2 = {}, acc3 = {};
    const float* aRow = h   + (size_t)(m0 + lo) * DD + 2 * hi;
    const float* bRow = Wfc + (size_t)(j0 + lo) * DD + 2 * hi;

    for (int k0 = 0; k0 < DD; k0 += 4) {
        v2f a  = *(const v2f*)(aRow + k0);
        v2f b0 = *(const v2f*)(bRow + k0);
        v2f b1 = *(const v2f*)(bRow + 16 * DD + k0);
        v2f b2 = *(const v2f*)(bRow + 32 * DD + k0);
        v2f b3 = *(const v2f*)(bRow + 48 * DD + k0);
        acc0 = wmma4(a, b0, acc0);
        acc1 = wmma4(a, b1, acc1);
        acc2 = wmma4(a, b2, acc2);
        acc3 = wmma4(a, b3, acc3);
    }
#pragma unroll
    for (int i = 0; i < 8; ++i) {
        const int row = m0 + i + 8 * hi;
        float* zr = z + (size_t)row * DD + j0 + lo;
        zr[0]  = acc0[i];
        zr[16] = acc1[i];
        zr[32] = acc2[i];
        zr[48] = acc3[i];
    }
}

// ---------------------------------------------------------------------------
// Fused per-node dots: e_src = z.a[:D], e_dst = z.a[D:], p0 = z.w_row0, p1 = z.w_row1
// one wave per node, 8 elements per lane, shfl_xor reduction
// ---------------------------------------------------------------------------
__global__ __launch_bounds__(256) void k_dots(const float* __restrict__ z,
                                              const float* __restrict__ a_attn,
                                              const float* __restrict__ w_row,
                                              float* __restrict__ es, float* __restrict__ ed,
                                              float* __restrict__ p0, float* __restrict__ p1) {
    const int n    = (blockIdx.x * blockDim.x + threadIdx.x) >> 5;
    const int lane = threadIdx.x & 31;
    const float* zr = z + (size_t)n * DD + lane * 8;

    float4 zA = *(const float4*)(zr);
    float4 zB = *(const float4*)(zr + 4);

    const float* aS = a_attn + lane * 8;
    const float* aD = a_attn + DD + lane * 8;
    const float* w0 = w_row + lane * 8;
    const float* w1 = w_row + DD + lane * 8;

    float4 s0 = *(const float4*)(aS), s1 = *(const float4*)(aS + 4);
    float4 d0 = *(const float4*)(aD), d1 = *(const float4*)(aD + 4);
    float4 r0 = *(const float4*)(w0), r1 = *(const float4*)(w0 + 4);
    float4 q0 = *(const float4*)(w1), q1 = *(const float4*)(w1 + 4);

    float vs = zA.x*s0.x + zA.y*s0.y + zA.z*s0.z + zA.w*s0.w
             + zB.x*s1.x + zB.y*s1.y + zB.z*s1.z + zB.w*s1.w;
    float vd = zA.x*d0.x + zA.y*d0.y + zA.z*d0.z + zA.w*d0.w
             + zB.x*d1.x + zB.y*d1.y + zB.z*d1.z + zB.w*d1.w;
    float v0 = zA.x*r0.x + zA.y*r0.y + zA.z*r0.z + zA.w*r0.w
             + zB.x*r1.x + zB.y*r1.y + zB.z*r1.z + zB.w*r1.w;
    float v1 = zA.x*q0.x + zA.y*q0.y + zA.z*q0.z + zA.w*q0.w
             + zB.x*q1.x + zB.y*q1.y + zB.z*q1.z + zB.w*q1.w;

#pragma unroll
    for (int m = 16; m >= 1; m >>= 1) {
        vs += __shfl_xor(vs, m, 32);
        vd += __shfl_xor(vd, m, 32);
        v0 += __shfl_xor(v0, m, 32);
        v1 += __shfl_xor(v1, m, 32);
    }
    if (lane == 0) { es[n] = vs; ed[n] = vd; p0[n] = v0; p1[n] = v1; }
}

// ---------------------------------------------------------------------------
// Per-node attention softmax + row/col conv + BN + ReLU -> G[N][272] (padded)
// one wave per node
// ---------------------------------------------------------------------------
__global__ __launch_bounds__(256) void k_node(const int* __restrict__ nbr,
                                              const float* __restrict__ z,
                                              const float* __restrict__ es,
                                              const float* __restrict__ ed,
                                              const float* __restrict__ p0,
                                              const float* __restrict__ p1,
                                              const float* __restrict__ w_col,
                                              const float* __restrict__ b_row,
                                              const float* __restrict__ g_row,
                                              const float* __restrict__ be_row,
                                              const float* __restrict__ b_col,
                                              const float* __restrict__ g_col,
                                              const float* __restrict__ be_col,
                                              float* __restrict__ G) {
    const int n    = (blockIdx.x * blockDim.x + threadIdx.x) >> 5;
    const int lane = threadIdx.x & 31;

    int   idx[KK];
    float e[KK];
    const float edn = ed[n];
    float mx = -3.4e38f;
#pragma unroll
    for (int k = 0; k < KK; ++k) {
        idx[k] = nbr[n * KK + k];
        float t = es[idx[k]] + edn;
        t = (t >= 0.0f) ? t : 0.01f * t;          // leaky_relu(0.01)
        e[k] = t;
        mx = fmaxf(mx, t);
    }
    float ssum = 0.0f;
#pragma unroll
    for (int k = 0; k < KK; ++k) { e[k] = expf(e[k] - mx); ssum += e[k]; }
    const float inv = 1.0f / ssum;

    const float bn_inv = rsqrtf(1.0f + BN_EPS);
    const float sc_col = g_col[0] * bn_inv, bo_col = be_col[0];
    const float sc_row = g_row[0] * bn_inv, bo_row = be_row[0];
    const float bcol = b_col[0], brow = b_row[0];

    // column conv: col[d] = sum_k (alpha_k * w_col[k]) * z[idx[k]][d]
    float cc[KK];
    const float* zp[KK];
#pragma unroll
    for (int k = 0; k < KK; ++k) {
        cc[k] = e[k] * inv * w_col[k];
        zp[k] = z + (size_t)idx[k] * DD;
    }
    float* Gr = G + (size_t)n * GSTR;
#pragma unroll
    for (int t = 0; t < DD / 32; ++t) {
        const int d = lane + t * 32;
        float acc = 0.0f;
#pragma unroll
        for (int k = 0; k < KK; ++k) acc += cc[k] * zp[k][d];
        acc = (acc + bcol) * sc_col + bo_col;
        Gr[9 + d] = fmaxf(acc, 0.0f);
    }

    // row conv: row_r = alpha_r*p0[idx_r] + alpha_{r+1}*p1[idx_{r+1}] + b_row
    float a_r = 0.0f, a_r1 = 0.0f, pp0 = 0.0f, pp1 = 0.0f;
#pragma unroll
    for (int k = 0; k < KK; ++k) {
        const float al = e[k] * inv;
        if (k == lane)     { a_r  = al; pp0 = p0[idx[k]]; }
        if (k == lane + 1) { a_r1 = al; pp1 = p1[idx[k]]; }
    }
    if (lane < 9) {
        float r = (a_r * pp0 + a_r1 * pp1 + brow) * sc_row + bo_row;
        Gr[lane] = fmaxf(r, 0.0f);
    } else if (lane < 16) {
        Gr[265 + (lane - 9)] = 0.0f;              // zero the pad columns
    }
}

// ---------------------------------------------------------------------------
// Build padded localw: Lp[272][256] (rows >= 265 zero)
// ---------------------------------------------------------------------------
__global__ __launch_bounds__(256) void k_lpad(const float* __restrict__ localw,
                                              float* __restrict__ Lp) {
    const int i = blockIdx.x * blockDim.x + threadIdx.x;
    if (i >= GSTR * DD) return;
    const int t = i / DD;
    Lp[i] = (t < 265) ? localw[i] : 0.0f;
}

// ---------------------------------------------------------------------------
// GEMM2: uf[n][j] = relu( sum_t G[n][t]*Lp[t][j] + h[n][j] )
// ---------------------------------------------------------------------------
__global__ __launch_bounds__(32) void k_gemm_uf(const float* __restrict__ G,
                                                const float* __restrict__ Lp,
                                                const float* __restrict__ h,
                                                float* __restrict__ uf) {
    const int m0   = blockIdx.x * 16;
    const int j0   = blockIdx.y * 64;
    const int lane = threadIdx.x;
    const int lo   = lane & 15;
    const int hi   = lane >> 4;

    v8f acc0 = {}, acc1 = {}, acc2 = {}, acc3 = {};
    const float* aRow = G + (size_t)(m0 + lo) * GSTR + 2 * hi;

    for (int k0 = 0; k0 < GSTR; k0 += 4) {
        v2f a = *(const v2f*)(aRow + k0);
        const int kr = k0 + 2 * hi;                   // B rows kr, kr+1 for this lane group
        const float* bp = Lp + (size_t)kr * DD + j0 + lo;
        v2f b0, b1, b2, b3;
        b0.x = bp[0];  b0.y = bp[DD];
        b1.x = bp[16]; b1.y = bp[DD + 16];
        b2.x = bp[32]; b2.y = bp[DD + 32];
        b3.x = bp[48]; b3.y = bp[DD + 48];
        acc0 = wmma4(a, b0, acc0);
        acc1 = wmma4(a, b1, acc1);
        acc2 = wmma4(a, b2, acc2);
        acc3 = wmma4(a, b3, acc3);
    }
#pragma unroll
    for (int i = 0; i < 8; ++i) {
        const int row = m0 + i + 8 * hi;
        const float* hr = h  + (size_t)row * DD + j0 + lo;
        float*       ur = uf + (size_t)row * DD + j0 + lo;
        ur[0]  = fmaxf(acc0[i] + hr[0],  0.0f);
        ur[16] = fmaxf(acc1[i] + hr[16], 0.0f);
        ur[32] = fmaxf(acc2[i] + hr[32], 0.0f);
        ur[48] = fmaxf(acc3[i] + hr[48], 0.0f);
    }
}

// ---------------------------------------------------------------------------
// GlobalAttention helpers
// ---------------------------------------------------------------------------
__global__ __launch_bounds__(256) void k_zero(float* __restrict__ p, int n) {
    const int i = blockIdx.x * blockDim.x + threadIdx.x;
    if (i < n) p[i] = 0.0f;
}

__global__ __launch_bounds__(256) void k_deg(const int* __restrict__ nbr,
                                             float* __restrict__ deg) {
    const int i = blockIdx.x * blockDim.x + threadIdx.x;
    if (i < NN * KK) atomicAdd(&deg[nbr[i]], 1.0f);
}

// hw[n] = dot(h[n], Wgc) * n_src[n]; one wave per node
__global__ __launch_bounds__(256) void k_hw(const float* __restrict__ h,
                                            const float* __restrict__ Wgc,
                                            const float* __restrict__ deg,
                                            float* __restrict__ hw) {
    const int n    = (blockIdx.x * blockDim.x + threadIdx.x) >> 5;
    const int lane = threadIdx.x & 31;
    const float* hr = h + (size_t)n * DD + lane * 8;
    const float* wr = Wgc + lane * 8;
    float4 hA = *(const float4*)(hr), hB = *(const float4*)(hr + 4);
    float4 wA = *(const float4*)(wr), wB = *(const float4*)(wr + 4);
    float v = hA.x*wA.x + hA.y*wA.y + hA.z*wA.z + hA.w*wA.w
            + hB.x*wB.x + hB.y*wB.y + hB.z*wB.z + hB.w*wB.w;
#pragma unroll
    for (int m = 16; m >= 1; m >>= 1) v += __shfl_xor(v, m, 32);
    if (lane == 0) {
        const float d = deg[n];
        const float ns = (d > 0.0f) ? rsqrtf(fmaxf(d, 1.0f)) : 0.0f;
        hw[n] = v * ns;
    }
}

// agg[n] = (sum_k hw[nbr[n][k]]) * rsqrt(K) + b_gc; softmax over groups of 256
__global__ __launch_bounds__(256) void k_aggw(const int* __restrict__ nbr,
                                              const float* __restrict__ hw,
                                              const float* __restrict__ b_gc,
                                              float* __restrict__ wgt) {
    __shared__ float red[256];
    const int tid = threadIdx.x;
    const int n   = blockIdx.x * 256 + tid;
    float a = 0.0f;
#pragma unroll
    for (int k = 0; k < KK; ++k) a += hw[nbr[n * KK + k]];
    a = a * rsqrtf((float)KK) + b_gc[0];

    red[tid] = a; __syncthreads();
#pragma unroll
    for (int s = 128; s > 0; s >>= 1) {
        if (tid < s) red[tid] = fmaxf(red[tid], red[tid + s]);
        __syncthreads();
    }
    const float mx = red[0]; __syncthreads();
    const float ex = expf(a - mx);
    red[tid] = ex; __syncthreads();
#pragma unroll
    for (int s = 128; s > 0; s >>= 1) {
        if (tid < s) red[tid] += red[tid + s];
        __syncthreads();
    }
    wgt[n] = ex / red[0];
}

// hg[d] += sum_{rows of this block} wgt[n]*uf[n][d]
__global__ __launch_bounds__(256) void k_red(const float* __restrict__ uf,
                                             const float* __restrict__ wgt,
                                             float* __restrict__ hg) {
    const int d    = threadIdx.x;
    const int base = blockIdx.x * 256;
    float s = 0.0f;
    for (int r = 0; r < 256; ++r) {
        const int n = base + r;
        s += wgt[n] * uf[(size_t)n * DD + d];
    }
    atomicAdd(&hg[d], s);
}

__global__ __launch_bounds__(64) void k_cls(const float* __restrict__ hg,
                                            const float* __restrict__ Wcls,
                                            const float* __restrict__ b_cls,
                                            float* __restrict__ out) {
    const int c = threadIdx.x;
    if (c >= CC) return;
    const float invN = 1.0f / (float)NN;
    float s = 0.0f;
    for (int d = 0; d < DD; ++d) s += (hg[d] * invN) * Wcls[d * CC + c];
    out[c] = s + b_cls[c];
}

// ---------------------------------------------------------------------------
extern "C" void kernel_launch(void* const* d_in, const int* in_sizes, int n_in,
                              void* d_out, int out_size, void* d_ws, size_t ws_size,
                              hipStream_t stream) {
    const float* h      = (const float*)d_in[0];
    const int*   nbr    = (const int*)  d_in[1];
    const float* Wfc    = (const float*)d_in[2];
    const float* a_attn = (const float*)d_in[3];
    const float* w_row  = (const float*)d_in[4];
    const float* b_row  = (const float*)d_in[5];
    const float* g_row  = (const float*)d_in[6];
    const float* be_row = (const float*)d_in[7];
    const float* w_col  = (const float*)d_in[8];
    const float* b_col  = (const float*)d_in[9];
    const float* g_col  = (const float*)d_in[10];
    const float* be_col = (const float*)d_in[11];
    const float* localw = (const float*)d_in[12];
    const float* Wgc    = (const float*)d_in[13];
    const float* b_gc   = (const float*)d_in[14];
    const float* Wcls   = (const float*)d_in[15];
    const float* bcls   = (const float*)d_in[16];
    float* out = (float*)d_out;

    // workspace layout (floats); zbuf is reused as updatefeat after G is built
    float* ws   = (float*)d_ws;
    float* zbuf = ws;                               // N*D   (z, later uf)
    float* G    = zbuf + (size_t)NN * DD;           // N*272
    float* es   = G    + (size_t)NN * GSTR;         // N
    float* ed   = es + NN;
    float* p0   = ed + NN;
    float* p1   = p0 + NN;
    float* deg  = p1 + NN;
    float* hw   = deg + NN;
    float* wgt  = hw + NN;
    float* Lp   = wgt + NN;                         // 272*256
    float* hg   = Lp + (size_t)GSTR * DD;           // 256

    // 1) z = h @ Wfc^T  (WMMA fp32)
    k_gemm_z<<<dim3(NN / 16, DD / 64), 32, 0, stream>>>(h, Wfc, zbuf);
    // 2) per-node dots
    k_dots<<<NN / 8, 256, 0, stream>>>(zbuf, a_attn, w_row, es, ed, p0, p1);
    // 3) attention + row/col conv -> G
    k_node<<<NN / 8, 256, 0, stream>>>(nbr, zbuf, es, ed, p0, p1, w_col,
                                       b_row, g_row, be_row, b_col, g_col, be_col, G);
    // 4) padded localw
    k_lpad<<<(GSTR * DD + 255) / 256, 256, 0, stream>>>(localw, Lp);
    // 5) uf = relu(G @ Lp + h)  (WMMA fp32; overwrites zbuf, z is dead)
    k_gemm_uf<<<dim3(NN / 16, DD / 64), 32, 0, stream>>>(G, Lp, h, zbuf);
    // 6) degree + hw
    k_zero<<<(NN + 255) / 256, 256, 0, stream>>>(deg, NN);
    k_deg<<<(NN * KK + 255) / 256, 256, 0, stream>>>(nbr, deg);
    k_hw<<<NN / 8, 256, 0, stream>>>(h, Wgc, deg, hw);
    // 7) group softmax weights
    k_aggw<<<NN / 256, 256, 0, stream>>>(nbr, hw, b_gc, wgt);
    // 8) weighted mean -> hg
    k_zero<<<1, 256, 0, stream>>>(hg, DD);
    k_red<<<NN / 256, 256, 0, stream>>>(zbuf, wgt, hg);
    // 9) classifier
    k_cls<<<1, 64, 0, stream>>>(hg, Wcls, bcls, out);
}